// Net2f_1254130450771
// MI455X (gfx1250) — compile-verified
//
#include <hip/hip_runtime.h>
#include <math.h>

#define NNODES 50000
#define NEDGES 800000
#define HF 256
#define NEG_SLOPE 0.2f

typedef __attribute__((ext_vector_type(2))) float v2f;
typedef __attribute__((ext_vector_type(8))) float v8f;

// Monotonic float<->uint order mapping (for deterministic atomic segment-max).
__device__ __forceinline__ unsigned f2ord(float f) {
    unsigned u = __float_as_uint(f);
    return (u & 0x80000000u) ? ~u : (u | 0x80000000u);
}
__device__ __forceinline__ float ord2f(unsigned u) {
    return __uint_as_float((u & 0x80000000u) ? (u ^ 0x80000000u) : ~u);
}

// ---------------------------------------------------------------------------
// C[n x 256] = act(A[n x 256]) @ W[256 x 256] (+ bias), fp32 WMMA 16x16x4.
// Block: 512 threads = 16 waves. Each block: one 16-row strip of A in LDS.
// Each wave: one 16x16 output tile (column tile = wave id).
// f32 WMMA lane layout (ISA 7.12.2):
//   A 16x4 : lanes 0-15 hold M=lane, v[0]=K0,v[1]=K1 ; lanes16-31: K2,K3
//   B 4x16 : lanes 0-15 hold N=lane, v[0]=K0,v[1]=K1 ; lanes16-31: K2,K3
//   C 16x16: vgpr r, lane L -> row r + 8*(L>=16), col L%16
// ---------------------------------------------------------------------------
__global__ __launch_bounds__(512)
void gemm256_wmma(const float* __restrict__ A, const float* __restrict__ W,
                  const float* __restrict__ bias, float* __restrict__ C,
                  int relu_in) {
    __shared__ float As[16][260];   // stride 260 -> 64-bank conflict-free column reads
    const int tid = threadIdx.x;
    const int rowbase = blockIdx.x * 16;

    // Cooperative load of the 16x256 A tile (coalesced), fused ReLU.
    #pragma unroll
    for (int i = 0; i < 8; ++i) {
        int e = tid + i * 512;
        int r = e >> 8, c = e & 255;
        float v = A[(size_t)(rowbase + r) * HF + c];
        if (relu_in) v = fmaxf(v, 0.0f);
        As[r][c] = v;
    }
    __syncthreads();

    const int wave = tid >> 5;      // 0..15 : column tile
    const int lane = tid & 31;
    const int half = lane >> 4;     // 0 or 1 (selects K pair)
    const int l16  = lane & 15;
    const int colbase = wave * 16;

    v8f acc = {0.f, 0.f, 0.f, 0.f, 0.f, 0.f, 0.f, 0.f};

    #pragma unroll 4
    for (int k = 0; k < HF; k += 4) {
        const int kk = k + 2 * half;
        v2f a, b;
        a.x = As[l16][kk];
        a.y = As[l16][kk + 1];
        b.x = W[(size_t)kk * HF + colbase + l16];
        b.y = W[(size_t)(kk + 1) * HF + colbase + l16];
        acc = __builtin_amdgcn_wmma_f32_16x16x4_f32(
            /*neg_a=*/false, a, /*neg_b=*/false, b,
            /*c_mod=*/(short)0, acc, /*reuse_a=*/false, /*reuse_b=*/false);
    }

    const int col = colbase + l16;
    const float bv = bias ? bias[col] : 0.0f;
    #pragma unroll
    for (int r = 0; r < 8; ++r) {
        int row = rowbase + r + 8 * half;
        C[(size_t)row * HF + col] = acc[r] + bv;
    }
}

// ---------------------------------------------------------------------------
// el[i] = feat[i,:] . al ; er[i] = feat[i,:] . ar     (wave per row)
// ---------------------------------------------------------------------------
__global__ __launch_bounds__(256)
void row_dots(const float* __restrict__ feat, const float* __restrict__ al,
              const float* __restrict__ ar, float* __restrict__ el,
              float* __restrict__ er) {
    const int wave = threadIdx.x >> 5;
    const int lane = threadIdx.x & 31;
    const int row = blockIdx.x * 8 + wave;
    if (row >= NNODES) return;
    const float* f = feat + (size_t)row * HF;
    float sl = 0.f, sr = 0.f;
    #pragma unroll
    for (int j = 0; j < 8; ++j) {
        int c = lane + j * 32;
        float v = f[c];
        sl += v * al[c];
        sr += v * ar[c];
    }
    #pragma unroll
    for (int off = 16; off > 0; off >>= 1) {
        sl += __shfl_xor(sl, off, 32);
        sr += __shfl_xor(sr, off, 32);
    }
    if (lane == 0) { el[row] = sl; er[row] = sr; }
}

// ---------------------------------------------------------------------------
// Init: h_out[i,c] = bias[c] (so the aggregation atomics land on the bias),
// maxb = 0 = minimal key (below ord(-inf)), denom = 0.
// ---------------------------------------------------------------------------
__global__ void init_layer(float* __restrict__ hout, const float* __restrict__ bias,
                           unsigned* __restrict__ maxb, float* __restrict__ denom) {
    const size_t idx = (size_t)blockIdx.x * blockDim.x + threadIdx.x;
    if (idx < (size_t)NNODES * HF) hout[idx] = bias[idx & 255];
    if (idx < NNODES) { maxb[idx] = 0u; denom[idx] = 0.f; }
}

__device__ __forceinline__ float edge_score(float a, float b) {
    float x = a + b;
    return x > 0.f ? x : NEG_SLOPE * x;
}

__global__ void edge_max(const int* __restrict__ src, const int* __restrict__ dst,
                         const float* __restrict__ el, const float* __restrict__ er,
                         unsigned* __restrict__ maxb) {
    const int e = blockIdx.x * blockDim.x + threadIdx.x;
    if (e >= NEDGES) return;
    const int d = dst[e];
    const float s = edge_score(el[src[e]], er[d]);
    atomicMax(&maxb[d], f2ord(s));
}

__global__ void edge_expsum(const int* __restrict__ src, const int* __restrict__ dst,
                            const float* __restrict__ el, const float* __restrict__ er,
                            const unsigned* __restrict__ maxb, float* __restrict__ denom) {
    const int e = blockIdx.x * blockDim.x + threadIdx.x;
    if (e >= NEDGES) return;
    const int d = dst[e];
    const float s = edge_score(el[src[e]], er[d]);
    atomicAdd(&denom[d], expf(s - ord2f(maxb[d])));
}

// ---------------------------------------------------------------------------
// h_out[dst,:] += a * feat[src,:]  — wave per edge, float4 gathers, f32 atomics.
// ---------------------------------------------------------------------------
__global__ __launch_bounds__(256)
void edge_agg(const int* __restrict__ src, const int* __restrict__ dst,
              const float* __restrict__ el, const float* __restrict__ er,
              const unsigned* __restrict__ maxb, const float* __restrict__ denom,
              const float* __restrict__ feat, float* __restrict__ hout) {
    const int wave = threadIdx.x >> 5;
    const int lane = threadIdx.x & 31;
    const long e = (long)blockIdx.x * 8 + wave;
    if (e >= NEDGES) return;
    const int s = src[e], d = dst[e];
    const float x = edge_score(el[s], er[d]);
    const float a = expf(x - ord2f(maxb[d])) / denom[d];
    const float4* fs = (const float4*)(feat + (size_t)s * HF);
    float* ho = hout + (size_t)d * HF;
    #pragma unroll
    for (int h = 0; h < 2; ++h) {
        const int c4 = lane + h * 32;       // float4 index within the 256-wide row
        const float4 v = fs[c4];
        const int c = c4 * 4;
        atomicAdd(ho + c + 0, v.x * a);
        atomicAdd(ho + c + 1, v.y * a);
        atomicAdd(ho + c + 2, v.z * a);
        atomicAdd(ho + c + 3, v.w * a);
    }
}

// ---------------------------------------------------------------------------
// out[i,:] = sigmoid( concat(h1,h2,h3)[i,:] @ W_ro + b_ro )   (wave per row)
// ---------------------------------------------------------------------------
__global__ __launch_bounds__(256)
void readout(const float* __restrict__ h1, const float* __restrict__ h2,
             const float* __restrict__ h3, const float* __restrict__ Wro,
             const float* __restrict__ bro, float* __restrict__ out) {
    const int wave = threadIdx.x >> 5;
    const int lane = threadIdx.x & 31;
    const int row = blockIdx.x * 8 + wave;
    if (row >= NNODES) return;
    float acc[8] = {0.f,0.f,0.f,0.f,0.f,0.f,0.f,0.f};
    const float* hs[3] = { h1 + (size_t)row * HF, h2 + (size_t)row * HF, h3 + (size_t)row * HF };
    for (int seg = 0; seg < 3; ++seg) {
        const float* h = hs[seg];
        #pragma unroll
        for (int j = 0; j < 8; ++j) {
            const int c = lane + j * 32;
            const float v = h[c];
            const float* w = Wro + (size_t)(seg * 256 + c) * 8;
            #pragma unroll
            for (int t = 0; t < 8; ++t) acc[t] += v * w[t];
        }
    }
    #pragma unroll
    for (int off = 16; off > 0; off >>= 1) {
        #pragma unroll
        for (int t = 0; t < 8; ++t) acc[t] += __shfl_xor(acc[t], off, 32);
    }
    if (lane == 0) {
        #pragma unroll
        for (int t = 0; t < 8; ++t)
            out[(size_t)row * 8 + t] = 1.0f / (1.0f + expf(-(acc[t] + bro[t])));
    }
}

extern "C" void kernel_launch(void* const* d_in, const int* in_sizes, int n_in,
                              void* d_out, int out_size, void* d_ws, size_t ws_size,
                              hipStream_t stream) {
    (void)in_sizes; (void)n_in; (void)out_size; (void)ws_size;

    const float* hv   = (const float*)d_in[0];
    const int*   src  = (const int*)d_in[1];
    const int*   dst  = (const int*)d_in[2];
    const float* W_in = (const float*)d_in[3];
    const float* b_in = (const float*)d_in[4];
    const float* W_ro = (const float*)d_in[5];
    const float* b_ro = (const float*)d_in[6];
    const float* fc[3]   = {(const float*)d_in[7],  (const float*)d_in[11], (const float*)d_in[15]};
    const float* al[3]   = {(const float*)d_in[8],  (const float*)d_in[12], (const float*)d_in[16]};
    const float* ar[3]   = {(const float*)d_in[9],  (const float*)d_in[13], (const float*)d_in[17]};
    const float* bias[3] = {(const float*)d_in[10], (const float*)d_in[14], (const float*)d_in[18]};

    float* ws = (float*)d_ws;
    const size_t NH = (size_t)NNODES * HF;
    float* h0    = ws;
    float* feat  = ws + NH;
    float* hL[3] = { ws + 2 * NH, ws + 3 * NH, ws + 4 * NH };
    float* el    = ws + 5 * NH;
    float* er    = el + NNODES;
    unsigned* maxb = (unsigned*)(er + NNODES);
    float* denom = (float*)(maxb + NNODES);

    const dim3 gemmGrid(NNODES / 16), gemmBlk(512);

    // h0 = hv @ W_in + b_in   (no input ReLU)
    gemm256_wmma<<<gemmGrid, gemmBlk, 0, stream>>>(hv, W_in, b_in, h0, 0);

    const float* hprev = h0;
    for (int l = 0; l < 3; ++l) {
        // feat = relu(hprev) @ fc
        gemm256_wmma<<<gemmGrid, gemmBlk, 0, stream>>>(hprev, fc[l], nullptr, feat, 1);
        row_dots<<<NNODES / 8, 256, 0, stream>>>(feat, al[l], ar[l], el, er);
        init_layer<<<(NNODES * HF + 255) / 256, 256, 0, stream>>>(hL[l], bias[l], maxb, denom);
        edge_max<<<(NEDGES + 255) / 256, 256, 0, stream>>>(src, dst, el, er, maxb);
        edge_expsum<<<(NEDGES + 255) / 256, 256, 0, stream>>>(src, dst, el, er, maxb, denom);
        edge_agg<<<NEDGES / 8, 256, 0, stream>>>(src, dst, el, er, maxb, denom, feat, hL[l]);
        hprev = hL[l];
    }

    readout<<<NNODES / 8, 256, 0, stream>>>(hL[0], hL[1], hL[2], W_ro, b_ro, (float*)d_out);
}